// InnerProduct_14620068675921
// MI455X (gfx1250) — compile-verified
//
#include <hip/hip_runtime.h>
#include <hip/hip_bf16.h>
#include <math.h>

typedef __attribute__((ext_vector_type(2))) float v2f;
typedef __attribute__((ext_vector_type(4))) float v4f;
typedef __attribute__((ext_vector_type(8))) float v8f;

#define D_FEAT 128

// One wave32 handles 16 edges via V_WMMA_F32_16X16X4_F32; diag(D) = dot products.
// K-axis permutation: lanes 0-15 stream features 0..63, lanes 16-31 features
// 64..127, so each lane feeds TWO consecutive WMMA steps from ONE b128 load.
__global__ void __launch_bounds__(256)
edge_dot_wmma(const float* __restrict__ z,
              const long long* __restrict__ erow,
              const long long* __restrict__ ecol,
              float* __restrict__ out,
              long long E) {
    const int lane = threadIdx.x & 31;
    const int wave = threadIdx.x >> 5;
    const long long e0 =
        ((long long)blockIdx.x * (blockDim.x >> 5) + wave) * 16;
    if (e0 >= E) return;                      // wave-uniform: EXEC stays all-1s

    const int m = lane & 15;                  // edge within group / matrix M or N
    long long e = e0 + m;
    if (e >= E) e = E - 1;                    // clamp (no divergence before WMMA)

    // Indices streamed exactly once -> non-temporal
    const long long r = __builtin_nontemporal_load(erow + e);
    const long long c = __builtin_nontemporal_load(ecol + e);

    // Half-wave feature split: lanes 0-15 -> feats [0,64), lanes 16-31 -> [64,128)
    const int half = lane >> 4;
    const float* pa = z + r * D_FEAT + half * 64;
    const float* pb = z + c * D_FEAT + half * 64;

    v8f acc = {};
#pragma unroll
    for (int t = 0; t < 16; ++t) {
        v4f a4 = *(const v4f*)(pa + 4 * t);   // global_load_b128, hits 192MB L2
        v4f b4 = *(const v4f*)(pb + 4 * t);
        v2f a01 = {a4.x, a4.y}, a23 = {a4.z, a4.w};
        v2f b01 = {b4.x, b4.y}, b23 = {b4.z, b4.w};
        acc = __builtin_amdgcn_wmma_f32_16x16x4_f32(
            false, a01, false, b01, (short)0, acc, false, false);
        acc = __builtin_amdgcn_wmma_f32_16x16x4_f32(
            false, a23, false, b23, (short)0, acc, false, false);
    }

    // Diagonal extraction:
    //   diag e in [0,8):  lane = e,      VGPR = e
    //   diag e in [8,16): lane = e + 16, VGPR = e - 8
    const int idx = (lane < 8) ? lane : ((lane >= 24) ? (lane - 24) : 0);
    float d = acc[0];
    d = (idx == 1) ? acc[1] : d;
    d = (idx == 2) ? acc[2] : d;
    d = (idx == 3) ? acc[3] : d;
    d = (idx == 4) ? acc[4] : d;
    d = (idx == 5) ? acc[5] : d;
    d = (idx == 6) ? acc[6] : d;
    d = (idx == 7) ? acc[7] : d;

    if (lane < 8 || lane >= 24) {
        const long long eo = e0 + ((lane < 8) ? lane : (lane - 16));
        if (eo < E) {
            const float sig = 1.0f / (1.0f + __expf(-d));
            __builtin_nontemporal_store(sig, out + eo);  // streamed once
        }
    }
}

extern "C" void kernel_launch(void* const* d_in, const int* in_sizes, int n_in,
                              void* d_out, int out_size, void* d_ws, size_t ws_size,
                              hipStream_t stream) {
    const float* z = (const float*)d_in[0];
    const long long* edge_index = (const long long*)d_in[1];  // int64 per reference
    float* out = (float*)d_out;

    const long long E = (long long)in_sizes[1] / 2;  // edge_index is (2, E)
    const long long* erow = edge_index;              // row-major: row block first
    const long long* ecol = edge_index + E;

    // 256 threads = 8 waves/block, 16 edges/wave -> 128 edges/block
    const int edges_per_block = 128;
    const int grid = (int)((E + edges_per_block - 1) / edges_per_block);
    edge_dot_wmma<<<grid, 256, 0, stream>>>(z, erow, ecol, out, E);
}